// ConditionedPNA_15341623181929
// MI455X (gfx1250) — compile-verified
//
#include <hip/hip_runtime.h>
#include <math.h>

// ---------------- problem constants ----------------
#define NN   10000          // nodes per copy
#define EE   50000          // directed edges
#define BB   4              // batch copies
#define KK   16             // negatives + 1
#define DD   64             // hidden dim
#define RR   200            // relations
#define LL   3              // layers
#define E2   (2*EE)         // undirected edges per copy  = 100000
#define NT   (BB*NN)        // total nodes                = 40000
#define ET   (BB*E2)        // total edges                = 400000
#define KS_N 4000           // int(0.1 * NT)
#define ES   40000          // int(1.0 * KS_N * ET / NT)

typedef float v2f __attribute__((ext_vector_type(2)));
typedef float v8f __attribute__((ext_vector_type(8)));

// order-preserving float<->uint mapping (no NaNs in this workload)
__device__ __forceinline__ unsigned ordkey(float f) {
    unsigned u = __float_as_uint(f);
    return (u & 0x80000000u) ? ~u : (u | 0x80000000u);
}
__device__ __forceinline__ float ordinv(unsigned k) {
    unsigned u = (k & 0x80000000u) ? (k ^ 0x80000000u) : ~k;
    return __uint_as_float(u);
}

// ---- CDNA5 async Global->LDS copy (ASYNCcnt-tracked DMA into LDS) ----
// LDS address is a per-lane VGPR offset; 16B per lane.
__device__ __forceinline__ void async_copy_b128(void* lds_ptr, const float* gptr) {
    unsigned lds_off = (unsigned)(size_t)lds_ptr;          // low 32 bits = LDS offset
    unsigned long long ga = (unsigned long long)(size_t)gptr;
    asm volatile("global_load_async_to_lds_b128 %0, %1, off"
                 :: "v"(lds_off), "v"(ga) : "memory");
}
__device__ __forceinline__ void wait_async0() {
    asm volatile("s_wait_asynccnt 0x0" ::: "memory");
}

// ================= init kernels =================

__global__ void init_zero_pre(float* score, int* degfull) {
    int i = blockIdx.x * blockDim.x + threadIdx.x;
    if (i < NT) score[i] = 0.0f;
    if (i < NN) degfull[i] = 0;
}

__global__ void init_hidden(const float* text, float* hidden) {
    int i = blockIdx.x * blockDim.x + threadIdx.x;   // over NT*DD
    if (i >= NT * DD) return;
    int node = i >> 6;
    int c    = i & 63;
    hidden[i] = (node < NN) ? text[node * DD + c] : 0.0f;
}

__global__ void degfull_accum(const int* ei, int* degfull) {
    int e = blockIdx.x * blockDim.x + threadIdx.x;   // over 2E
    if (e >= E2) return;
    int d = (e < EE) ? ei[EE + e] : ei[e - EE];      // reversed edges: dst = orig src
    atomicAdd(&degfull[d], 1);
}

__global__ void pna_reduce(const int* degfull, float* pna) {
    __shared__ float red[256];
    int t = threadIdx.x;
    float s = 0.0f;
    for (int i = t; i < NN; i += 256) s += logf((float)degfull[i] + 1.0f);
    red[t] = s;
    __syncthreads();
    for (int off = 128; off > 0; off >>= 1) {
        if (t < off) red[t] += red[t + off];
        __syncthreads();
    }
    if (t == 0) pna[0] = red[0] / (float)NN;         // mean over Nt == mean over N
}

__global__ void qlin_kernel(const int* r_index, const float* relTab,
                            const float* Wlin, const float* blin, float* qlin) {
    int n = threadIdx.x;                              // 64 threads
    if (n >= DD) return;
    int r0 = r_index[0];                              // query0 = rel_embeds[0]
    float acc = blin[n];
    for (int k = 0; k < DD; ++k)
        acc += relTab[(size_t)r0 * DD + k] * Wlin[(size_t)(DD + k) * DD + n];
    qlin[n] = acc;
}

// initial head embedding + initial score at the B head nodes
__global__ void head_init(const int* h_index, const int* r_index, const float* hs,
                          const float* relTab, const float* Wlin, const float* blin,
                          const float* W1, const float* b1, const float* W2,
                          const float* b2, float* hidden, float* score) {
    __shared__ float s_h[DD], s_r[DD], s_x[DD], s_m[2 * DD];
    int b = blockIdx.x;                               // grid = BB
    int t = threadIdx.x;                              // block = 128
    int h0 = h_index[b * KK] + b * NN;
    int r0 = r_index[b * KK];
    if (t < DD) {
        s_h[t] = hs[b * DD + t];
        s_r[t] = relTab[(size_t)r0 * DD + t];
    }
    __syncthreads();
    if (t < DD) {
        hidden[(size_t)h0 * DD + t] = s_h[t];
        float acc = blin[t];
        for (int k = 0; k < DD; ++k)
            acc += s_h[k] * Wlin[k * DD + t] + s_r[k] * Wlin[(DD + k) * DD + t];
        s_x[t] = s_h[t] * acc;                        // hidden * heuristic
    }
    __syncthreads();
    {
        float hv = b1[t];
        for (int k = 0; k < DD; ++k) hv += s_x[k] * W1[k * (2 * DD) + t];
        s_m[t] = hv > 0.0f ? hv : 0.0f;
    }
    __syncthreads();
    if (t == 0) {
        float sc = b2[0];
        for (int j = 0; j < 2 * DD; ++j) sc += s_m[j] * W2[j];
        score[h0] = sc;
    }
}

// ================= per-layer kernels =================

__global__ void layer_init(float* sum, float* sumsq, unsigned* maxb, unsigned* minb,
                           int* deg, int* outdeg) {
    int i = blockIdx.x * blockDim.x + threadIdx.x;   // over NT*DD
    if (i >= NT * DD) return;
    sum[i]   = 0.0f;
    sumsq[i] = 0.0f;
    maxb[i]  = 0u;                                    // minimal ordkey
    minb[i]  = 0xFFFFFFFFu;                           // maximal ordkey
    if (i < NT) { deg[i] = 0; outdeg[i] = 0; }
}

// k-th largest of score[] via bisection on ordkey space (exact, deterministic)
__global__ __launch_bounds__(1024)
void select_node_thr(const float* score, unsigned* thrOut) {
    __shared__ unsigned s_lo, s_hi;
    __shared__ int s_cnt;
    int t = threadIdx.x;
    if (t == 0) { s_lo = 0u; s_hi = 0xFFFFFFFFu; }
    __syncthreads();
    for (int it = 0; it < 40; ++it) {
        unsigned lo = s_lo, hi = s_hi;
        if (lo >= hi) break;                          // uniform (shared) -> safe
        unsigned mid = (unsigned)((((unsigned long long)lo + (unsigned long long)hi) + 1ull) >> 1);
        if (t == 0) s_cnt = 0;
        __syncthreads();
        int c = 0;
        for (int i = t; i < NT; i += 1024)
            if (ordkey(score[i]) >= mid) c++;
        atomicAdd(&s_cnt, c);
        __syncthreads();
        if (t == 0) { if (s_cnt >= KS_N) s_lo = mid; else s_hi = mid - 1u; }
        __syncthreads();
    }
    if (t == 0) thrOut[0] = s_lo;
}

// es-th largest of edge_score (score[src] if >= thr else -inf); masked edges -> key 0
__global__ __launch_bounds__(1024)
void select_edge_thr(const int* ei, const float* score, const unsigned* thrN,
                     unsigned* thrOut) {
    __shared__ unsigned s_lo, s_hi;
    __shared__ int s_cnt;
    int t = threadIdx.x;
    unsigned tn = thrN[0];
    if (t == 0) { s_lo = 0u; s_hi = 0xFFFFFFFFu; }
    __syncthreads();
    for (int it = 0; it < 40; ++it) {
        unsigned lo = s_lo, hi = s_hi;
        if (lo >= hi) break;
        unsigned mid = (unsigned)((((unsigned long long)lo + (unsigned long long)hi) + 1ull) >> 1);
        if (t == 0) s_cnt = 0;
        __syncthreads();
        int c = 0;
        for (int e = t; e < ET; e += 1024) {
            int b = e / E2;
            int r = e - b * E2;
            int s = (r < EE) ? ei[r] : ei[EE + (r - EE)];
            unsigned key = ordkey(score[s + b * NN]);
            if (key >= tn && key >= mid) c++;         // effkey = key if >=thr else 0
        }
        atomicAdd(&s_cnt, c);
        __syncthreads();
        if (t == 0) { if (s_cnt >= ES) s_lo = mid; else s_hi = mid - 1u; }
        __syncthreads();
    }
    if (t == 0) thrOut[0] = s_lo;
}

// gated relation-conditioned messages + PNA segment reductions via atomics
__global__ __launch_bounds__(256)
void message_kernel(const int* ei, const int* ea, const float* score,
                    const float* hidden, const float* relw,
                    const unsigned* thrN, const unsigned* thrE,
                    float* sum, float* sumsq, unsigned* maxb, unsigned* minb,
                    int* deg, int* outdeg) {
    int t = threadIdx.x;
    int e = blockIdx.x * 4 + (t >> 6);                // 4 edges / block, 64 ch each
    int c = t & 63;
    if (e >= ET) return;
    int b = e / E2;
    int r = e - b * E2;
    int s, d, attr;
    if (r < EE) { s = ei[r];            d = ei[EE + r];  attr = ea[r]; }
    else        { int rr = r - EE; s = ei[EE + rr]; d = ei[rr]; attr = ea[rr]; }
    int src = s + b * NN, dst = d + b * NN;
    float sv = score[src];
    unsigned key = ordkey(sv);
    if (key < thrN[0] || key < thrE[0]) return;       // emask
    float gate = 1.0f / (1.0f + expf(-sv));
    float m = gate * hidden[(size_t)src * DD + c] * relw[(size_t)attr * DD + c];
    atomicAdd(&sum  [(size_t)dst * DD + c], m);
    atomicAdd(&sumsq[(size_t)dst * DD + c], m * m);
    unsigned mk = ordkey(m);
    atomicMax(&maxb[(size_t)dst * DD + c], mk);
    atomicMin(&minb[(size_t)dst * DD + c], mk);
    if (c == 0) { atomicAdd(&deg[dst], 1); atomicAdd(&outdeg[src], 1); }
}

// feat[node, 0:64]=mean, 64:128=max, 128:192=min, 192:256=std
__global__ void finalize_feat(const float* sum, const float* sumsq,
                              const unsigned* maxb, const unsigned* minb,
                              const int* deg, float* feat) {
    int i = blockIdx.x * blockDim.x + threadIdx.x;    // over NT*DD
    if (i >= NT * DD) return;
    int node = i >> 6, c = i & 63;
    int dgi = deg[node];
    float dg = (float)dgi;
    float denom = dg > 1.0f ? dg : 1.0f;
    float mean = sum[i] / denom;
    float sq   = sumsq[i] / denom;
    float var  = sq - mean * mean;
    if (var < 0.0f) var = 0.0f;
    float stdv = sqrtf(var + 1e-6f);
    float mx = 0.0f, mn = 0.0f;
    if (dgi > 0) { mx = ordinv(maxb[i]); mn = ordinv(minb[i]); }
    size_t base = (size_t)node * (4 * DD) + c;
    feat[base]            = mean;
    feat[base + DD]       = mx;
    feat[base + 2 * DD]   = mn;
    feat[base + 3 * DD]   = stdv;
}

// ---------- fp32 WMMA (V_WMMA_F32_16X16X4_F32) with B from LDS ----------
__device__ __forceinline__ v8f wmma_lds(const float* Arow, const float* sB,
                                        int ldb, int k, int k0, int col, v8f acc) {
    v2f a, b;
    a.x = Arow[k + k0];
    a.y = Arow[k + k0 + 1];
    b.x = sB[(k + k0) * ldb + col];       // padded stride -> conflict-free ds_load
    b.y = sB[(k + k0 + 1) * ldb + col];
    return __builtin_amdgcn_wmma_f32_16x16x4_f32(false, a, false, b,
                                                 (short)0, acc, false, false);
}

// upd = relu(feat@W0 + amp*(feat@W1) + att*(feat@W2) + conv_b)
// block: 8 waves = 2 row-tiles x 4 col-tiles -> 32 rows x 64 cols per block.
// Weights staged per 64-row K-chunk into LDS via async Global->LDS DMA.
#define CSTRIDE 80                            // 64 + 16 pad: rows 2 apart -> 32 banks apart
__global__ __launch_bounds__(256)
void conv_gemm(const float* feat, const float* convW, const float* convb,
               const int* deg, const float* pna, float* upd) {
    __shared__ float sB[3 * 64 * CSTRIDE];    // 60 KB
    int tid  = threadIdx.x;
    int wave = tid >> 5, lane = tid & 31;
    int mtile = blockIdx.x * 2 + (wave >> 2);
    int ntile = wave & 3;
    int m = lane & 15, n = lane & 15, k0 = (lane >> 4) * 2;
    int col = ntile * 16 + n;
    const float* Arow = feat + ((size_t)(mtile * 16) + m) * (4 * DD);
    int rr = tid >> 4, q = tid & 15;          // staging: 16 lanes x 16 rows / pass
    v8f acc0 = {0,0,0,0,0,0,0,0};
    v8f acc1 = {0,0,0,0,0,0,0,0};
    v8f acc2 = {0,0,0,0,0,0,0,0};
    for (int kc = 0; kc < 4; ++kc) {          // K = 4 chunks of 64
        __syncthreads();                       // previous chunk fully consumed
#pragma unroll
        for (int mat = 0; mat < 3; ++mat)
#pragma unroll
            for (int p = 0; p < 4; ++p) {
                int row = p * 16 + rr;
                const float* g = convW + ((size_t)(mat * 256 + kc * 64 + row)) * DD + q * 4;
                async_copy_b128(&sB[(mat * 64 + row) * CSTRIDE + q * 4], g);
            }
        if (kc < 3)                            // prefetch next A chunk through GL2
            __builtin_prefetch(Arow + (kc + 1) * 64, 0, 3);
        wait_async0();
        __syncthreads();
#pragma unroll 4
        for (int k = 0; k < 64; k += 4) {
            acc0 = wmma_lds(Arow + kc * 64, sB,                    CSTRIDE, k, k0, col, acc0);
            acc1 = wmma_lds(Arow + kc * 64, sB + 64 * CSTRIDE,     CSTRIDE, k, k0, col, acc1);
            acc2 = wmma_lds(Arow + kc * 64, sB + 2 * 64 * CSTRIDE, CSTRIDE, k, k0, col, acc2);
        }
    }
    float pm = pna[0];
    float bias = convb[col];
    int rowbase = mtile * 16 + ((lane >> 4) * 8);
#pragma unroll
    for (int i = 0; i < 8; ++i) {
        int row = rowbase + i;
        float dgl = logf((float)deg[row] + 1.0f);
        float amp = dgl / pm;
        float att = pm / (dgl + 1e-6f);
        float v = acc0[i] + amp * acc1[i] + att * acc2[i] + bias;
        upd[(size_t)row * DD + col] = v > 0.0f ? v : 0.0f;
    }
}

__global__ void update_hidden(const float* upd, const int* outdeg, float* hidden) {
    int i = blockIdx.x * blockDim.x + threadIdx.x;    // over NT*DD
    if (i >= NT * DD) return;
    if (outdeg[i >> 6] > 0) hidden[i] += upd[i];
}

// x = hidden * (hidden @ W_lin[0:64] + qlin); W staged once into LDS
__global__ __launch_bounds__(256)
void heur_gemm(const float* hidden, const float* Wlin, const float* qlin, float* xbuf) {
    __shared__ float sW[64 * CSTRIDE];         // 20 KB
    int tid  = threadIdx.x;
    int wave = tid >> 5, lane = tid & 31;
    int mtile = blockIdx.x * 2 + (wave >> 2);
    int ntile = wave & 3;
    int m = lane & 15, n = lane & 15, k0 = (lane >> 4) * 2;
    int col = ntile * 16 + n;
    int rr = tid >> 4, q = tid & 15;
#pragma unroll
    for (int p = 0; p < 4; ++p) {
        int row = p * 16 + rr;
        async_copy_b128(&sW[row * CSTRIDE + q * 4], Wlin + (size_t)row * DD + q * 4);
    }
    const float* Arow = hidden + ((size_t)(mtile * 16) + m) * DD;
    wait_async0();
    __syncthreads();
    v8f acc = {0,0,0,0,0,0,0,0};
#pragma unroll
    for (int k = 0; k < DD; k += 4)
        acc = wmma_lds(Arow, sW, CSTRIDE, k, k0, col, acc);
    float qv = qlin[col];
    int rowbase = mtile * 16 + ((lane >> 4) * 8);
#pragma unroll
    for (int i = 0; i < 8; ++i) {
        int row = rowbase + i;
        float heur = acc[i] + qv;
        xbuf[(size_t)row * DD + col] = hidden[(size_t)row * DD + col] * heur;
    }
}

// h = relu(x @ W_mlp1 + b1)   (N=128 -> 8 col tiles, 16 rows per block)
#define MSTRIDE 144                            // 128 + 16 pad
__global__ __launch_bounds__(256)
void mlp1_gemm(const float* xbuf, const float* W1, const float* b1, float* hbuf) {
    __shared__ float sW[64 * MSTRIDE];         // 36 KB
    int tid  = threadIdx.x;
    int wave = tid >> 5, lane = tid & 31;
    int mtile = blockIdx.x;
    int ntile = wave;                           // 0..7
    int m = lane & 15, n = lane & 15, k0 = (lane >> 4) * 2;
    int col = ntile * 16 + n;
    int rr = tid >> 5, q = tid & 31;            // 32 lanes x 8 rows / pass
#pragma unroll
    for (int p = 0; p < 8; ++p) {
        int row = p * 8 + rr;
        async_copy_b128(&sW[row * MSTRIDE + q * 4], W1 + (size_t)row * (2 * DD) + q * 4);
    }
    const float* Arow = xbuf + ((size_t)(mtile * 16) + m) * DD;
    wait_async0();
    __syncthreads();
    v8f acc = {0,0,0,0,0,0,0,0};
#pragma unroll
    for (int k = 0; k < DD; k += 4)
        acc = wmma_lds(Arow, sW, MSTRIDE, k, k0, col, acc);
    float bias = b1[col];
    int rowbase = mtile * 16 + ((lane >> 4) * 8);
#pragma unroll
    for (int i = 0; i < 8; ++i) {
        int row = rowbase + i;
        float v = acc[i] + bias;
        hbuf[(size_t)row * (2 * DD) + col] = v > 0.0f ? v : 0.0f;
    }
}

__global__ void score_update(const float* hbuf, const float* W2, const float* b2,
                             const int* outdeg, float* score) {
    int node = blockIdx.x * blockDim.x + threadIdx.x;
    if (node >= NT) return;
    const float* h = hbuf + (size_t)node * (2 * DD);
    float s = b2[0];
#pragma unroll 8
    for (int j = 0; j < 2 * DD; ++j) s += h[j] * W2[j];
    if (outdeg[node] > 0) score[node] = s;
}

__global__ void gather_out(const int* t_index, const float* score, float* out) {
    int t = threadIdx.x;                               // 64 threads
    if (t >= BB * KK) return;
    int b = t >> 4;
    out[t] = score[t_index[t] + b * NN];
}

// ================= launch =================

extern "C" void kernel_launch(void* const* d_in, const int* in_sizes, int n_in,
                              void* d_out, int out_size, void* d_ws, size_t ws_size,
                              hipStream_t stream) {
    const int*   h_index  = (const int*)  d_in[0];
    const int*   r_index  = (const int*)  d_in[1];
    const int*   t_index  = (const int*)  d_in[2];
    const float* hs       = (const float*)d_in[3];
    const int*   ei       = (const int*)  d_in[5];
    const int*   ea       = (const int*)  d_in[6];
    const float* text     = (const float*)d_in[7];
    const float* relTab   = (const float*)d_in[9];
    const float* Wlin     = (const float*)d_in[10];
    const float* blin     = (const float*)d_in[11];
    const float* W1       = (const float*)d_in[12];
    const float* b1       = (const float*)d_in[13];
    const float* W2       = (const float*)d_in[14];
    const float* b2       = (const float*)d_in[15];
    const float* rel_w    = (const float*)d_in[16];
    const float* conv_W   = (const float*)d_in[17];
    const float* conv_b   = (const float*)d_in[18];
    float* out = (float*)d_out;

    // ---- workspace carve-up (4-byte units) ----
    float* W = (float*)d_ws;
    size_t o = 0;
    float*    score   = W + o;              o += NT;
    float*    hidden  = W + o;              o += (size_t)NT * DD;
    float*    sumb    = W + o;              o += (size_t)NT * DD;   // alias: upd
    float*    sumsq   = W + o;              o += (size_t)NT * DD;   // alias: xbuf
    unsigned* maxb    = (unsigned*)(W + o); o += (size_t)NT * DD;   // alias: hbuf lo
    unsigned* minb    = (unsigned*)(W + o); o += (size_t)NT * DD;   // alias: hbuf hi
    int*      deg     = (int*)(W + o);      o += NT;
    int*      outdeg  = (int*)(W + o);      o += NT;
    float*    feat    = W + o;              o += (size_t)NT * 4 * DD;
    int*      degfull = (int*)(W + o);      o += NN;
    float*    pna     = W + o;              o += 1;
    unsigned* thrN    = (unsigned*)(W + o); o += 1;
    unsigned* thrE    = (unsigned*)(W + o); o += 1;
    float*    qlin    = W + o;              o += DD;
    float*    updb    = sumb;                                        // aliases
    float*    xbuf    = sumsq;
    float*    hbuf    = (float*)maxb;      // spans maxb+minb = NT*128 floats

    const int TPB = 256;
    const int nd_blocks = (NT * DD + TPB - 1) / TPB;                 // 10000

    // ---- graph-constant precompute ----
    init_zero_pre<<<(NT + TPB - 1) / TPB, TPB, 0, stream>>>(score, degfull);
    init_hidden<<<nd_blocks, TPB, 0, stream>>>(text, hidden);
    degfull_accum<<<(E2 + TPB - 1) / TPB, TPB, 0, stream>>>(ei, degfull);
    pna_reduce<<<1, 256, 0, stream>>>(degfull, pna);
    qlin_kernel<<<1, 64, 0, stream>>>(r_index, relTab, Wlin, blin, qlin);
    head_init<<<BB, 128, 0, stream>>>(h_index, r_index, hs, relTab, Wlin, blin,
                                      W1, b1, W2, b2, hidden, score);

    // ---- layers ----
    for (int l = 0; l < LL; ++l) {
        const float* relw_l  = rel_w  + (size_t)l * 2 * RR * DD;
        const float* convW_l = conv_W + (size_t)l * 12 * DD * DD;
        const float* convb_l = conv_b + (size_t)l * DD;

        layer_init<<<nd_blocks, TPB, 0, stream>>>(sumb, sumsq, maxb, minb, deg, outdeg);
        select_node_thr<<<1, 1024, 0, stream>>>(score, thrN);
        select_edge_thr<<<1, 1024, 0, stream>>>(ei, score, thrN, thrE);
        message_kernel<<<ET / 4, TPB, 0, stream>>>(ei, ea, score, hidden, relw_l,
                                                   thrN, thrE, sumb, sumsq,
                                                   maxb, minb, deg, outdeg);
        finalize_feat<<<nd_blocks, TPB, 0, stream>>>(sumb, sumsq, maxb, minb, deg, feat);
        conv_gemm<<<NT / 32, TPB, 0, stream>>>(feat, convW_l, convb_l, deg, pna, updb);
        update_hidden<<<nd_blocks, TPB, 0, stream>>>(updb, outdeg, hidden);
        heur_gemm<<<NT / 32, TPB, 0, stream>>>(hidden, Wlin, qlin, xbuf);
        mlp1_gemm<<<NT / 16, TPB, 0, stream>>>(xbuf, W1, b1, hbuf);
        score_update<<<(NT + TPB - 1) / TPB, TPB, 0, stream>>>(hbuf, W2, b2, outdeg, score);
    }

    gather_out<<<1, 64, 0, stream>>>(t_index, score, out);
}